// KinematicsLayer_9620726743069
// MI455X (gfx1250) — compile-verified
//
#include <hip/hip_runtime.h>
#include <hip/hip_bf16.h>

#define N_CHAINS 5
#define N_JOINTS 4
#define N_VERTS  512
#define BATCHN   4096
#define QDIM     (6 + N_CHAINS * N_JOINTS)   // 26
#define VCHUNK   128                          // verts per block-iteration (8 waves * 16)
#define LDS_BSTRIDE 388                       // 128*3 = 384 + 4 pad: kills bank conflicts, keeps 16B align

typedef __attribute__((ext_vector_type(2))) float v2f;
typedef __attribute__((ext_vector_type(4))) float v4f;
typedef __attribute__((ext_vector_type(8))) float v8f;

// Rodrigues: R = I + s*K + (1-c)*K^2 for unit axis u
__device__ __forceinline__ void rot_from_axis(float ux, float uy, float uz,
                                              float s, float c, float R[3][3]) {
    float c1 = 1.0f - c;
    R[0][0] = c + ux * ux * c1;      R[0][1] = ux * uy * c1 - uz * s; R[0][2] = ux * uz * c1 + uy * s;
    R[1][0] = uy * ux * c1 + uz * s; R[1][1] = c + uy * uy * c1;      R[1][2] = uy * uz * c1 - ux * s;
    R[2][0] = uz * ux * c1 - uy * s; R[2][1] = uz * uy * c1 + ux * s; R[2][2] = c + uz * uz * c1;
}

// ---------------------------------------------------------------------------
// Phase 1: forward kinematics. One thread per (b, c) chain. Writes row-major
// 4x4 transform per (b, c, j) into workspace (81920 * 16 floats = 5.24 MB).
// ---------------------------------------------------------------------------
__global__ void fk_kernel(const float* __restrict__ qpos,
                          const float* __restrict__ offsets,
                          const float* __restrict__ axes,
                          float* __restrict__ ws) {
    int idx = blockIdx.x * blockDim.x + threadIdx.x;
    if (idx >= BATCHN * N_CHAINS) return;
    int b = idx / N_CHAINS;
    int c = idx - b * N_CHAINS;

    const float* qp = qpos + (size_t)b * QDIM;

    // Global pose: hom(axis_angle_to_matrix(aa), pos)
    float a0 = qp[3], a1 = qp[4], a2 = qp[5];
    float ang = sqrtf(a0 * a0 + a1 * a1 + a2 * a2);
    float inv = 1.0f / fmaxf(ang, 1e-8f);
    float Rg[3][3];
    rot_from_axis(a0 * inv, a1 * inv, a2 * inv, sinf(ang), cosf(ang), Rg);

    float T[3][4];   // bottom row [0,0,0,1] implicit
    #pragma unroll
    for (int r = 0; r < 3; ++r) {
        T[r][0] = Rg[r][0]; T[r][1] = Rg[r][1]; T[r][2] = Rg[r][2];
        T[r][3] = qp[r];
    }

    #pragma unroll
    for (int j = 0; j < N_JOINTS; ++j) {
        float q = qp[6 + c * N_JOINTS + j];
        const float* ax = axes + (size_t)(c * N_JOINTS + j) * 3;
        float Rj[3][3];
        rot_from_axis(ax[0], ax[1], ax[2], sinf(q), cosf(q), Rj);

        const float* O = offsets + (size_t)(c * N_JOINTS + j) * 16;
        // M = O @ hom(Rj, 0); O bottom row is [0,0,0,1] so M bottom row stays [0,0,0,1]
        float M[3][4];
        #pragma unroll
        for (int r = 0; r < 3; ++r) {
            #pragma unroll
            for (int k = 0; k < 3; ++k)
                M[r][k] = O[r * 4 + 0] * Rj[0][k] + O[r * 4 + 1] * Rj[1][k] + O[r * 4 + 2] * Rj[2][k];
            M[r][3] = O[r * 4 + 3];
        }
        // T = T @ M  (M row 3 = [0,0,0,1])
        float Tn[3][4];
        #pragma unroll
        for (int r = 0; r < 3; ++r) {
            #pragma unroll
            for (int k = 0; k < 3; ++k)
                Tn[r][k] = T[r][0] * M[0][k] + T[r][1] * M[1][k] + T[r][2] * M[2][k];
            Tn[r][3] = T[r][0] * M[0][3] + T[r][1] * M[1][3] + T[r][2] * M[2][3] + T[r][3];
        }
        #pragma unroll
        for (int r = 0; r < 3; ++r)
            #pragma unroll
            for (int k = 0; k < 4; ++k)
                T[r][k] = Tn[r][k];

        float* w = ws + ((size_t)(b * N_CHAINS + c) * N_JOINTS + j) * 16;
        #pragma unroll
        for (int r = 0; r < 3; ++r) {
            w[r * 4 + 0] = T[r][0]; w[r * 4 + 1] = T[r][1];
            w[r * 4 + 2] = T[r][2]; w[r * 4 + 3] = T[r][3];
        }
        w[12] = 0.0f; w[13] = 0.0f; w[14] = 0.0f; w[15] = 1.0f;
    }
}

// ---------------------------------------------------------------------------
// Phase 2: vertex transform via V_WMMA_F32_16X16X4_F32.
//   A (16x4)  = 16 verts as rows [x y z 1]     (lanes 0-15: K0/K1, 16-31: K2/K3)
//   B (4x16)  = col (4*bb + comp) = row `comp` of transform T[b0+bb, c, j]
//   D (16x16) = comp `n%4` of vert `m` for batch element b0 + n/4
// Epilogue stages D through LDS, then issues coalesced 128-bit NT stores with
// all address math hoisted out of the (fully unrolled) iteration loop.
// Block = 256 threads (8 waves); block owns one (c, j) and 4 batch elements.
// ---------------------------------------------------------------------------
__global__ void __launch_bounds__(256)
vert_kernel(const float* __restrict__ verts,
            const float* __restrict__ ws,
            float* __restrict__ out) {
    __shared__ float lds[4 * LDS_BSTRIDE];   // 6208 bytes

    int blk  = blockIdx.x;
    int bgrp = blk / (N_CHAINS * N_JOINTS);
    int rem  = blk - bgrp * (N_CHAINS * N_JOINTS);
    int c    = rem / N_JOINTS;
    int j    = rem - c * N_JOINTS;
    int b0   = bgrp * 4;

    int tid  = threadIdx.x;
    int wave = tid >> 5;
    int lane = tid & 31;
    int half = lane >> 4;     // 0 => this lane holds K=0,1 ; 1 => K=2,3
    int n    = lane & 15;     // matrix column / row-in-lane index
    int bb   = n >> 2;        // which of the 4 batch elements
    int comp = n & 3;         // output component (3 == discarded w)

    // B matrix (loop invariant): two consecutive elements of transform row `comp`
    const float* Tw = ws + (((size_t)(b0 + bb) * N_CHAINS + c) * N_JOINTS + j) * 16
                         + comp * 4 + half * 2;
    v2f bmat;
    bmat.x = Tw[0];
    bmat.y = Tw[1];

    // A matrix base pointers (loop adds only the compile-time stride it*384)
    const float* vbase = verts + (size_t)(c * N_JOINTS + j) * N_VERTS * 3;
    const float* vp  = vbase + (size_t)(wave * 16 + n) * 3;  // this lane's vertex
    const float* vpa = vp + half * 2;                        // x (half=0) or z (half=1)

    // LDS scatter base (loop invariant)
    int lds_w = bb * LDS_BSTRIDE + (wave * 16 + half * 8) * 3 + comp;

    // Epilogue: thread t owns float4 slot t, and slot t+256 iff t < 128.
    // Decompose g -> (batch bb2, run r2) ONCE; per iteration only +it*VCHUNK*3.
    int bb0s = tid / 96;
    int r0s  = tid - bb0s * 96;
    const float* lsrc0 = &lds[bb0s * LDS_BSTRIDE + r0s * 4];
    float* odst0 = out + (((size_t)(b0 + bb0s) * N_CHAINS + c) * N_JOINTS + j)
                             * (N_VERTS * 3) + r0s * 4;
    bool has2 = tid < 128;
    int g1   = tid + 256;
    int bb1s = g1 / 96;
    int r1s  = g1 - bb1s * 96;
    const float* lsrc1 = &lds[bb1s * LDS_BSTRIDE + r1s * 4];
    float* odst1 = out + (((size_t)(b0 + bb1s) * N_CHAINS + c) * N_JOINTS + j)
                             * (N_VERTS * 3) + r1s * 4;

    #pragma unroll
    for (int it = 0; it < N_VERTS / VCHUNK; ++it) {
        // ---- A matrix: branchless loads (vp[1] is always in-bounds) ----
        float fa = vpa[it * (VCHUNK * 3)];       // x or z
        float fy = vp[it * (VCHUNK * 3) + 1];    // y
        v2f amat;
        amat.x = fa;
        amat.y = half ? 1.0f : fy;               // v_cndmask, no EXEC divergence

        v8f acc = {};
        acc = __builtin_amdgcn_wmma_f32_16x16x4_f32(
            /*neg_a=*/false, amat, /*neg_b=*/false, bmat,
            /*c_mod=*/(short)0, acc, /*reuse_a=*/false, /*reuse_b=*/false);

        // ---- Scatter D into LDS (comp==3 lanes carry w -> dropped) ----
        if (comp < 3) {
            #pragma unroll
            for (int r = 0; r < 8; ++r)
                lds[lds_w + r * 3] = acc[r];
        }
        __syncthreads();

        // ---- Coalesced 128-bit NT stores; only immediate offsets vary ----
        {
            v4f val = *(const v4f*)lsrc0;
            __builtin_nontemporal_store(val, (v4f*)(odst0 + it * (VCHUNK * 3)));
        }
        if (has2) {
            v4f val = *(const v4f*)lsrc1;
            __builtin_nontemporal_store(val, (v4f*)(odst1 + it * (VCHUNK * 3)));
        }
        __syncthreads();
    }
}

extern "C" void kernel_launch(void* const* d_in, const int* in_sizes, int n_in,
                              void* d_out, int out_size, void* d_ws, size_t ws_size,
                              hipStream_t stream) {
    const float* qpos    = (const float*)d_in[0];   // (4096, 26)
    const float* offsets = (const float*)d_in[1];   // (5, 4, 4, 4)
    const float* axes    = (const float*)d_in[2];   // (5, 4, 3)
    const float* verts   = (const float*)d_in[3];   // (5, 4, 512, 3)
    float* out = (float*)d_out;                     // (4096, 5, 4, 512, 3)
    float* ws  = (float*)d_ws;                      // needs 81920*16*4 = 5.24 MB

    // Phase 1: one thread per (b, c) chain.
    int chains = BATCHN * N_CHAINS;                 // 20480
    fk_kernel<<<(chains + 255) / 256, 256, 0, stream>>>(qpos, offsets, axes, ws);

    // Phase 2: one block per (4 batch elements, c, j).
    int blocks = (BATCHN / 4) * N_CHAINS * N_JOINTS; // 20480
    vert_kernel<<<blocks, 256, 0, stream>>>(verts, ws, out);
}